// MambaBlock_60060822667885
// MI455X (gfx1250) — compile-verified
//
#include <hip/hip_runtime.h>
#include <hip/hip_bf16.h>

typedef float v2f __attribute__((ext_vector_type(2)));
typedef float v8f __attribute__((ext_vector_type(8)));

#define N_EMBD   768
#define D_INNER  1536
#define D_STATE  64
#define SEQ      1024
#define BATCH    4
#define M_ROWS   (BATCH * SEQ)   // 4096

// ---------------------------------------------------------------------------
// Generic f32 GEMM:  C[M,N] = X[M,K] @ W[N,K]^T   (row-major X, W, C)
// One wave computes a 32(M) x 64(N) tile: 8x V_WMMA_F32_16X16X4_F32 per
// K-step of 4, fed by 6 contiguous float2 loads (2x A, 4x B).
// Software-pipelined (prefetch next K-step into registers before current
// WMMAs) and addressed via per-lane pointers (named scalars, not arrays)
// so loads take the global_load + immediate-offset form.
// ---------------------------------------------------------------------------
__global__ __launch_bounds__(256) void gemm_f32_wmma(
    const float* __restrict__ X, const float* __restrict__ W,
    float* __restrict__ C, int M, int N, int K)
{
    const int lane  = threadIdx.x & 31;
    const int wave  = threadIdx.x >> 5;
    const int tilesM = M >> 5;                 // M multiple of 32
    const int tilesN = (N + 63) >> 6;
    const int tIdx = blockIdx.x * 8 + wave;
    if (tIdx >= tilesM * tilesN) return;
    // consecutive waves share the same N tile -> W loads hit L0/L2
    const int tN = tIdx / tilesM;
    const int tM = tIdx % tilesM;
    const int m0 = tM << 5;
    const int n0 = tN << 6;

    const int lrow  = lane & 15;               // matrix row within 16-tile
    const int khalf = (lane >> 4) << 1;        // 0 or 2 (K pair select)

    const float* pA0 = X + (size_t)(m0 + lrow) * K + khalf;
    const float* pA1 = pA0 + (size_t)16 * K;
    int r0 = n0 +  0 + lrow; if (r0 > N - 1) r0 = N - 1;
    int r1 = n0 + 16 + lrow; if (r1 > N - 1) r1 = N - 1;
    int r2 = n0 + 32 + lrow; if (r2 > N - 1) r2 = N - 1;
    int r3 = n0 + 48 + lrow; if (r3 > N - 1) r3 = N - 1;
    const float* pB0 = W + (size_t)r0 * K + khalf;
    const float* pB1 = W + (size_t)r1 * K + khalf;
    const float* pB2 = W + (size_t)r2 * K + khalf;
    const float* pB3 = W + (size_t)r3 * K + khalf;

    v8f acc[8] = {};   // [mSub*4 + nSub]

    // ---- prologue: load K-step 0 operands
    v2f a0 = *(const v2f*)pA0;
    v2f a1 = *(const v2f*)pA1;
    v2f b0 = *(const v2f*)pB0;
    v2f b1 = *(const v2f*)pB1;
    v2f b2 = *(const v2f*)pB2;
    v2f b3 = *(const v2f*)pB3;

    for (int k = 0; k + 4 < K; k += 4) {
        // prefetch next K-step (immediate offset) while current WMMAs run
        v2f na0 = *(const v2f*)(pA0 + 4);
        v2f na1 = *(const v2f*)(pA1 + 4);
        v2f nb0 = *(const v2f*)(pB0 + 4);
        v2f nb1 = *(const v2f*)(pB1 + 4);
        v2f nb2 = *(const v2f*)(pB2 + 4);
        v2f nb3 = *(const v2f*)(pB3 + 4);
        pA0 += 4; pA1 += 4; pB0 += 4; pB1 += 4; pB2 += 4; pB3 += 4;

        acc[0] = __builtin_amdgcn_wmma_f32_16x16x4_f32(false, a0, false, b0, (short)0, acc[0], false, false);
        acc[1] = __builtin_amdgcn_wmma_f32_16x16x4_f32(false, a0, false, b1, (short)0, acc[1], false, false);
        acc[2] = __builtin_amdgcn_wmma_f32_16x16x4_f32(false, a0, false, b2, (short)0, acc[2], false, false);
        acc[3] = __builtin_amdgcn_wmma_f32_16x16x4_f32(false, a0, false, b3, (short)0, acc[3], false, false);
        acc[4] = __builtin_amdgcn_wmma_f32_16x16x4_f32(false, a1, false, b0, (short)0, acc[4], false, false);
        acc[5] = __builtin_amdgcn_wmma_f32_16x16x4_f32(false, a1, false, b1, (short)0, acc[5], false, false);
        acc[6] = __builtin_amdgcn_wmma_f32_16x16x4_f32(false, a1, false, b2, (short)0, acc[6], false, false);
        acc[7] = __builtin_amdgcn_wmma_f32_16x16x4_f32(false, a1, false, b3, (short)0, acc[7], false, false);

        a0 = na0; a1 = na1; b0 = nb0; b1 = nb1; b2 = nb2; b3 = nb3;
    }
    // ---- epilogue: final K-step
    acc[0] = __builtin_amdgcn_wmma_f32_16x16x4_f32(false, a0, false, b0, (short)0, acc[0], false, false);
    acc[1] = __builtin_amdgcn_wmma_f32_16x16x4_f32(false, a0, false, b1, (short)0, acc[1], false, false);
    acc[2] = __builtin_amdgcn_wmma_f32_16x16x4_f32(false, a0, false, b2, (short)0, acc[2], false, false);
    acc[3] = __builtin_amdgcn_wmma_f32_16x16x4_f32(false, a0, false, b3, (short)0, acc[3], false, false);
    acc[4] = __builtin_amdgcn_wmma_f32_16x16x4_f32(false, a1, false, b0, (short)0, acc[4], false, false);
    acc[5] = __builtin_amdgcn_wmma_f32_16x16x4_f32(false, a1, false, b1, (short)0, acc[5], false, false);
    acc[6] = __builtin_amdgcn_wmma_f32_16x16x4_f32(false, a1, false, b2, (short)0, acc[6], false, false);
    acc[7] = __builtin_amdgcn_wmma_f32_16x16x4_f32(false, a1, false, b3, (short)0, acc[7], false, false);

    // C/D layout: VGPR r -> row r (lanes 0-15) / row r+8 (lanes 16-31),
    // column = lane & 15
#pragma unroll
    for (int i = 0; i < 2; ++i) {
        const int mbase = m0 + 16 * i + ((lane >> 4) << 3);
#pragma unroll
        for (int j = 0; j < 4; ++j) {
            const int col = n0 + j * 16 + lrow;
            if (col < N) {
#pragma unroll
                for (int r = 0; r < 8; ++r)
                    C[(size_t)(mbase + r) * N + col] = acc[i * 4 + j][r];
            }
        }
    }
}

// ---------------------------------------------------------------------------
// Depthwise causal conv (width 4) + SiLU over xz[:, :D_INNER]
// ---------------------------------------------------------------------------
__global__ __launch_bounds__(256) void conv_silu_kernel(
    const float* __restrict__ xz, const float* __restrict__ cw,
    const float* __restrict__ cb, float* __restrict__ out)
{
    const int idx = blockIdx.x * 256 + threadIdx.x;   // < M_ROWS * D_INNER
    const int d = idx % D_INNER;
    const int m = idx / D_INNER;
    const int t = m & (SEQ - 1);
    float acc = cb[d];
    const float* wp = cw + d * 4;                     // conv_w[d,0,0..3]
#pragma unroll
    for (int j = 0; j < 4; ++j) {
        const int tt = t - 3 + j;
        if (tt >= 0)
            acc += wp[j] * xz[(size_t)(m - 3 + j) * (2 * D_INNER) + d];
    }
    out[idx] = acc / (1.0f + __expf(-acc));           // SiLU
}

// ---------------------------------------------------------------------------
// Selective scan.  One wave per (b,d); lane owns states {2*lane, 2*lane+1}.
// Fuses: delta = softplus(x_dbl[:, :4] @ W_dt[d,:]^T + b_dt[d]),
// recurrence, y_t = sum_n h*C (shfl_xor reduce), + D*x, * silu(res).
// Result overwrites xconv in place (each slot read-then-written by its
// owning wave within the same timestep).
// ---------------------------------------------------------------------------
#define XDBL_LD 132   // DT_RANK(4) + 2*D_STATE(128)

__global__ __launch_bounds__(256) void scan_kernel(
    const float* __restrict__ xz, float* xconv_io,
    const float* __restrict__ xdbl, const float* __restrict__ Wdt,
    const float* __restrict__ bdt, const float* __restrict__ Alog,
    const float* __restrict__ Dvec)
{
    const int lane = threadIdx.x & 31;
    const int i = blockIdx.x * 8 + (threadIdx.x >> 5);  // (b,d) flat
    const int b = i / D_INNER;
    const int d = i % D_INNER;

    const float A0 = -__expf(Alog[d * D_STATE + 2 * lane]);
    const float A1 = -__expf(Alog[d * D_STATE + 2 * lane + 1]);
    const float w0 = Wdt[d * 4 + 0], w1 = Wdt[d * 4 + 1];
    const float w2 = Wdt[d * 4 + 2], w3 = Wdt[d * 4 + 3];
    const float bb = bdt[d];
    const float Dd = Dvec[d];

    float h0 = 0.f, h1 = 0.f;
    for (int t = 0; t < SEQ; ++t) {
        const int m = b * SEQ + t;
        const float4 xq = *(const float4*)(xdbl + (size_t)m * XDBL_LD);
        const float draw = xq.x * w0 + xq.y * w1 + xq.z * w2 + xq.w * w3 + bb;
        const float delta = (draw > 20.f) ? draw : log1pf(__expf(draw));
        const float xv = xconv_io[(size_t)m * D_INNER + d];
        const float2 Bv = *(const float2*)(xdbl + (size_t)m * XDBL_LD + 4 + 2 * lane);
        const float2 Cv = *(const float2*)(xdbl + (size_t)m * XDBL_LD + 4 + D_STATE + 2 * lane);
        const float dx = delta * xv;
        h0 = __expf(delta * A0) * h0 + dx * Bv.x;
        h1 = __expf(delta * A1) * h1 + dx * Bv.y;
        float p = h0 * Cv.x + h1 * Cv.y;
#pragma unroll
        for (int off = 16; off > 0; off >>= 1)
            p += __shfl_xor(p, off, 32);
        if (lane == 0) {
            const float res = xz[(size_t)m * (2 * D_INNER) + D_INNER + d];
            const float y = p + Dd * xv;
            xconv_io[(size_t)m * D_INNER + d] = y * (res / (1.f + __expf(-res)));
        }
    }
}

// ---------------------------------------------------------------------------
extern "C" void kernel_launch(void* const* d_in, const int* in_sizes, int n_in,
                              void* d_out, int out_size, void* d_ws, size_t ws_size,
                              hipStream_t stream)
{
    (void)in_sizes; (void)n_in; (void)out_size; (void)ws_size;
    const float* x      = (const float*)d_in[0];
    const float* W_in   = (const float*)d_in[1];
    const float* conv_w = (const float*)d_in[2];
    const float* conv_b = (const float*)d_in[3];
    const float* W_x    = (const float*)d_in[4];
    const float* W_dt   = (const float*)d_in[5];
    const float* b_dt   = (const float*)d_in[6];
    const float* A_log  = (const float*)d_in[7];
    const float* Dv     = (const float*)d_in[8];
    const float* W_out  = (const float*)d_in[9];
    float* out = (float*)d_out;

    char* ws = (char*)d_ws;
    const size_t xz_bytes    = (size_t)M_ROWS * 2 * D_INNER * 4;  // 50.3 MB
    const size_t xconv_bytes = (size_t)M_ROWS * D_INNER * 4;      // 25.2 MB
    float* xz    = (float*)ws;
    float* xconv = (float*)(ws + xz_bytes);
    float* xdbl  = (float*)(ws + xz_bytes + xconv_bytes);         // 2.2 MB

    // 1) xr = x @ W_in^T   (M=4096, N=3072, K=768)
    {
        const int tiles = (M_ROWS / 32) * ((2 * D_INNER + 63) / 64);
        gemm_f32_wmma<<<(tiles + 7) / 8, 256, 0, stream>>>(
            x, W_in, xz, M_ROWS, 2 * D_INNER, N_EMBD);
    }
    // 2) depthwise conv + SiLU
    conv_silu_kernel<<<(M_ROWS * D_INNER) / 256, 256, 0, stream>>>(
        xz, conv_w, conv_b, xconv);
    // 3) x_dbl = x_ssm @ W_x^T   (M=4096, N=132, K=1536)
    {
        const int tiles = (M_ROWS / 32) * ((XDBL_LD + 63) / 64);
        gemm_f32_wmma<<<(tiles + 7) / 8, 256, 0, stream>>>(
            xconv, W_x, xdbl, M_ROWS, XDBL_LD, D_INNER);
    }
    // 4) selective scan (fused delta / D*x / silu(res) gating); in-place
    scan_kernel<<<(BATCH * D_INNER) / 8, 256, 0, stream>>>(
        xz, xconv, xdbl, W_dt, b_dt, A_log, Dv);
    // 5) out = ypre @ W_out^T   (M=4096, N=768, K=1536)
    {
        const int tiles = (M_ROWS / 32) * (N_EMBD / 64);
        gemm_f32_wmma<<<(tiles + 7) / 8, 256, 0, stream>>>(
            xconv, W_out, out, M_ROWS, N_EMBD, D_INNER);
    }
}